// DeformConv_10943576670962
// MI455X (gfx1250) — compile-verified
//
#include <hip/hip_runtime.h>
#include <hip/hip_bf16.h>

typedef float v2f __attribute__((ext_vector_type(2)));
typedef float v8f __attribute__((ext_vector_type(8)));

#define Bn   4
#define Cn   64
#define Tn   8
#define Hn   128
#define Wn   128
#define DGn  8
#define CGn  8                      // channels per deformable group
#define THWn (Tn * Hn * Wn)         // 131072
#define NPTS (Bn * Tn * Hn * Wn)    // 524288 spatial points

#define WAVES_PER_BLOCK 4
#define TILE_P 16                   // spatial points per wave

__global__ __launch_bounds__(128)
void deform_conv3d_wmma_kernel(const float* __restrict__ x,
                               const float* __restrict__ off,
                               const float* __restrict__ wgt,
                               const float* __restrict__ bias,
                               float* __restrict__ out)
{
    // Weight 64x64, row-padded to 65 floats: A-frag reads are bank-conflict free
    __shared__ float Wl[Cn][Cn + 1];
    // Per-wave sampled tile S[c][p], row-padded to 17 floats
    __shared__ float Sl[WAVES_PER_BLOCK][Cn][TILE_P + 1];

    const int tid  = threadIdx.x;
    const int lane = tid & 31;
    const int wave = tid >> 5;

    // Cooperative weight load (shared by all 4 waves)
    for (int i = tid; i < Cn * Cn; i += blockDim.x)
        Wl[i >> 6][i & 63] = wgt[i];

    const int tile  = blockIdx.x * WAVES_PER_BLOCK + wave;
    const int pbase = tile * TILE_P;

    // ---------- Phase 1: deformable trilinear sampling into LDS ----------
    // 16 points x 8 groups = 128 (point,group) pairs per wave -> 4 per lane.
    #pragma unroll
    for (int i = 0; i < 4; ++i) {
        const int q = lane + 32 * i;        // 0..127
        const int p = q & 15;               // point within tile
        const int g = q >> 4;               // deformable group
        const int P = pbase + p;
        const int w = P & (Wn - 1);
        const int h = (P >> 7) & (Hn - 1);
        const int t = (P >> 14) & (Tn - 1);
        const int b = P >> 17;

        const int offBase = (((b * (3 * DGn) + g * 3) * Tn + t) * Hn + h) * Wn + w;
        const float o0 = off[offBase];
        const float o1 = off[offBase + THWn];
        const float o2 = off[offBase + 2 * THWn];

        const float gt = (float)t + o0;
        const float gh = (float)h + o1;
        const float gw = (float)w + o2;
        const float t0f = floorf(gt), h0f = floorf(gh), w0f = floorf(gw);
        const float ft = gt - t0f, fh = gh - h0f, fw = gw - w0f;
        const int t0 = (int)t0f, h0 = (int)h0f, w0 = (int)w0f;

        float acc[CGn];
        #pragma unroll
        for (int c = 0; c < CGn; ++c) acc[c] = 0.0f;

        const float* xb = x + (size_t)(b * Cn + g * CGn) * THWn;

        #pragma unroll
        for (int corner = 0; corner < 8; ++corner) {
            const int dt = corner & 1;
            const int dh = (corner >> 1) & 1;
            const int dw = (corner >> 2) & 1;
            const int ti = t0 + dt, hi = h0 + dh, wi = w0 + dw;
            const bool valid = (ti >= 0) & (ti < Tn) & (hi >= 0) & (hi < Hn)
                             & (wi >= 0) & (wi < Wn);
            float wt = (dt ? ft : 1.0f - ft) * (dh ? fh : 1.0f - fh)
                     * (dw ? fw : 1.0f - fw);
            wt = valid ? wt : 0.0f;
            const int tc = min(max(ti, 0), Tn - 1);
            const int hc = min(max(hi, 0), Hn - 1);
            const int wc = min(max(wi, 0), Wn - 1);
            const int idx = (tc * Hn + hc) * Wn + wc;
            #pragma unroll
            for (int c = 0; c < CGn; ++c)
                acc[c] += wt * xb[(size_t)c * THWn + idx];
        }
        #pragma unroll
        for (int c = 0; c < CGn; ++c)
            Sl[wave][g * CGn + c][p] = acc[c];
    }

    __syncthreads();

    // ---------- Phase 2: 64x64 channel GEMM via V_WMMA_F32_16X16X4_F32 ----------
    // Per-lane frag coordinates (ISA 16x4 A / 4x16 B / 16x16 C layouts):
    //   lanes 0-15:  N/M = lane,    K pair = {k0, k0+1}
    //   lanes 16-31: N/M = lane-16, K pair = {k0+2, k0+3}
    const int n    = lane & 15;
    const int hilo = lane >> 4;

    const int P = pbase + n;             // this lane's output point (column)
    const int w = P & (Wn - 1);
    const int h = (P >> 7) & (Hn - 1);
    const int t = (P >> 14) & (Tn - 1);
    const int b = P >> 17;
    const int spatial = (t * Hn + h) * Wn + w;

    for (int mt = 0; mt < 4; ++mt) {         // 4 tiles of 16 output channels
        v8f acc = {0.f, 0.f, 0.f, 0.f, 0.f, 0.f, 0.f, 0.f};
        #pragma unroll
        for (int kk = 0; kk < 16; ++kk) {    // K = 64 in steps of 4
            const int k0 = kk * 4 + hilo * 2;
            const int M  = mt * 16 + n;
            v2f a, bf;
            a.x  = Wl[M][k0];
            a.y  = Wl[M][k0 + 1];
            bf.x = Sl[wave][k0][n];
            bf.y = Sl[wave][k0 + 1][n];
            acc = __builtin_amdgcn_wmma_f32_16x16x4_f32(
                      /*neg_a=*/false, a, /*neg_b=*/false, bf,
                      /*c_mod=*/(short)0, acc, /*reuse_a=*/false, /*reuse_b=*/false);
        }
        // D layout: VGPR r -> row mt*16 + r + hilo*8, column n
        #pragma unroll
        for (int r = 0; r < 8; ++r) {
            const int o = mt * 16 + r + hilo * 8;
            out[(size_t)(b * Cn + o) * THWn + spatial] = acc[r] + bias[o];
        }
    }
}

extern "C" void kernel_launch(void* const* d_in, const int* in_sizes, int n_in,
                              void* d_out, int out_size, void* d_ws, size_t ws_size,
                              hipStream_t stream) {
    (void)in_sizes; (void)n_in; (void)out_size; (void)d_ws; (void)ws_size;
    const float* x    = (const float*)d_in[0];
    const float* off  = (const float*)d_in[1];
    const float* wgt  = (const float*)d_in[2];
    const float* bias = (const float*)d_in[3];
    float* out = (float*)d_out;

    const int tiles  = NPTS / TILE_P;                 // 32768 waves of work
    const int blocks = tiles / WAVES_PER_BLOCK;       // 8192 blocks of 128 thr
    deform_conv3d_wmma_kernel<<<blocks, 128, 0, stream>>>(x, off, wgt, bias, out);
}